// MyNet_27642409517656
// MI455X (gfx1250) — compile-verified
//
#include <hip/hip_runtime.h>
#include <hip/hip_bf16.h>

// ---------------------------------------------------------------------------
// MyNet forward, algebraically fused so the 256->4096 MLP output layers never
// materialize.  All GEMMs run on V_WMMA_F32_16X16X4_F32 (full fp32 precision;
// post-fusion the problem is only ~7.5 GFLOPs so fp32 WMMA is the right
// precision/performance point on MI455X).
//
// Input flattening assumption (insertion order of setup_inputs() dict, leaves
// in order): [0]=v, [1]=fc0.w, [2]=fc0.b, then per layer i (base L=3+16*i):
//   L..L+5   phi_i:  w0[64,56] b0[64] w1[256,64] b1[256] w2[4096,256] b2[4096]
//   L+6..11  psi_i:  same shapes
//   L+12,13  w_i[32,32], b_i[32]
//   L+14,15  bn_i: gamma[32], beta[32]
// [67]=fc1.w[128,32], [68]=fc1.b[128], [69]=fc2.w[1,128], [70]=fc2.b[1]
// ---------------------------------------------------------------------------

#define BB   4
#define NN   4096
#define BN   (BB*NN)          // 16384 rows
#define IND  56
#define WW   32
#define RR   4

typedef __attribute__((ext_vector_type(2))) float v2f;
typedef __attribute__((ext_vector_type(8))) float v8f;

// Only meaningful to check on the DEVICE pass; the host pass does not declare
// amdgcn builtins and would trip the #error spuriously.
#if defined(__HIP_DEVICE_COMPILE__)
#if !__has_builtin(__builtin_amdgcn_wmma_f32_16x16x4_f32)
#error "wmma_f32_16x16x4_f32 builtin not available for gfx1250 device pass"
#endif
#endif

__device__ __forceinline__ v8f wmma4(v2f a, v2f b, v8f c) {
  // (neg_a, A, neg_b, B, c_mod, C, reuse_a, reuse_b)
  return __builtin_amdgcn_wmma_f32_16x16x4_f32(false, a, false, b, (short)0, c,
                                               false, false);
}

// A fragment: A row-major [>=16 rows, lda], rows row0..row0+15, K-slice k0..k0+3
__device__ __forceinline__ v2f frag_a(const float* __restrict__ A, int lda, int k0) {
  int lane = threadIdx.x & 31;
  int m  = lane & 15;
  int kk = k0 + ((lane >> 4) << 1);
  v2f a;
  a.x = A[(size_t)m * lda + kk];
  a.y = A[(size_t)m * lda + kk + 1];
  return a;
}

// B fragment from a row-major weight W[col, k] (i.e. B = W^T); Wt points at
// W + col0*ldk.
__device__ __forceinline__ v2f frag_bT(const float* __restrict__ Wt, int ldk, int k0) {
  int lane = threadIdx.x & 31;
  int n  = lane & 15;
  int kk = k0 + ((lane >> 4) << 1);
  v2f b;
  b.x = Wt[(size_t)n * ldk + kk];
  b.y = Wt[(size_t)n * ldk + kk + 1];
  return b;
}

// ---------------------------------------------------------------------------
// fc0: v32 = v @ fc0w^T + fc0b          [BN,56] -> [BN,32]
// wave per (row-tile, col-tile): 1024*2 = 2048 waves; block 256 -> grid 256
// ---------------------------------------------------------------------------
__global__ void k_fc0(const float* __restrict__ A, const float* __restrict__ W,
                      const float* __restrict__ bias, float* __restrict__ out) {
  int wv   = blockIdx.x * 8 + (threadIdx.x >> 5);
  int lane = threadIdx.x & 31;
  int rt = wv >> 1, ct = wv & 1;
  const float* Ar = A + (size_t)rt * 16 * IND;
  const float* Wt = W + (size_t)ct * 16 * IND;
  v8f acc = {};
  for (int k0 = 0; k0 < IND; k0 += 4)
    acc = wmma4(frag_a(Ar, IND, k0), frag_bT(Wt, IND, k0), acc);
  int n = lane & 15, mh = (lane >> 4) * 8;
  float bc = bias[ct * 16 + n];
  float* Dr = out + ((size_t)rt * 16) * WW + ct * 16;
#pragma unroll
  for (int r = 0; r < 8; ++r) Dr[(size_t)(mh + r) * WW + n] = acc[r] + bc;
}

// ---------------------------------------------------------------------------
// MLP trunk: h2 = relu(relu(a@W0^T+b0)@W1^T+b1)   [BN,56]->[BN,64]->[BN,256]
// wave per 16-row tile (1024 tiles); block 256 (8 waves) -> grid 128.
// h1 tile is restaged through LDS to become the next A operand.
// ---------------------------------------------------------------------------
#define H1LD 68   // padded leading dim to dodge LDS bank conflicts
__global__ void k_mlp(const float* __restrict__ A,
                      const float* __restrict__ W0, const float* __restrict__ b0,
                      const float* __restrict__ W1, const float* __restrict__ b1,
                      float* __restrict__ H2) {
  __shared__ float h1s[8][16 * H1LD];
  int wvl  = threadIdx.x >> 5;
  int lane = threadIdx.x & 31;
  int rt   = blockIdx.x * 8 + wvl;
  const float* Ar = A + (size_t)rt * 16 * IND;
  float* h1 = &h1s[wvl][0];
  int n = lane & 15, mh = (lane >> 4) * 8;

  // h1 = relu(a @ W0^T + b0): 4 column tiles, K=56
#pragma unroll
  for (int ct = 0; ct < 4; ++ct) {
    v8f acc = {};
    for (int k0 = 0; k0 < IND; k0 += 4)
      acc = wmma4(frag_a(Ar, IND, k0), frag_bT(W0 + (size_t)ct * 16 * IND, IND, k0), acc);
    float bc = b0[ct * 16 + n];
#pragma unroll
    for (int r = 0; r < 8; ++r)
      h1[(size_t)(mh + r) * H1LD + ct * 16 + n] = fmaxf(acc[r] + bc, 0.f);
  }
  __syncthreads();

  // h2 = relu(h1 @ W1^T + b1): 16 column tiles, K=64, A from LDS
  float* H2r = H2 + (size_t)rt * 16 * 256;
  for (int ct = 0; ct < 16; ++ct) {
    if (ct < 15) __builtin_prefetch(W1 + (size_t)(ct + 1) * 16 * 64, 0, 0);
    v8f acc = {};
    for (int k0 = 0; k0 < 64; k0 += 4)
      acc = wmma4(frag_a(h1, H1LD, k0), frag_bT(W1 + (size_t)ct * 16 * 64, 64, k0), acc);
    float bc = b1[ct * 16 + n];
#pragma unroll
    for (int r = 0; r < 8; ++r)
      H2r[(size_t)(mh + r) * 256 + ct * 16 + n] = fmaxf(acc[r] + bc, 0.f);
  }
}

// ---------------------------------------------------------------------------
// sv[b,i] = sum_n v32[b,n,i]          grid=B blocks of 1024 threads
// ---------------------------------------------------------------------------
__global__ void k_sv(const float* __restrict__ V32, float* __restrict__ sv) {
  __shared__ float red[1024];
  int b = blockIdx.x;
  int i = threadIdx.x & 31, g = threadIdx.x >> 5;
  float s = 0.f;
  for (int nn = g; nn < NN; nn += 32)
    s += V32[((size_t)b * NN + nn) * WW + i];
  red[threadIdx.x] = s;
  __syncthreads();
  if (threadIdx.x < 32) {
    float t = 0.f;
    for (int g2 = 0; g2 < 32; ++g2) t += red[g2 * 32 + threadIdx.x];
    sv[b * 32 + threadIdx.x] = t;
  }
}

// ---------------------------------------------------------------------------
// G[b,i,c] = sum_n h2psi[b,n,c] * v32[b,n,i]   (A = v32^T, B = h2psi, K = 4096)
// wave per (b, it in 0..1, ct in 0..15): 128 waves; block 128 -> grid 32
// ---------------------------------------------------------------------------
__global__ void k_G(const float* __restrict__ V32, const float* __restrict__ H2,
                    float* __restrict__ G) {
  int wv   = (blockIdx.x * blockDim.x + threadIdx.x) >> 5;
  int lane = threadIdx.x & 31;
  int ct = wv & 15, it = (wv >> 4) & 1, b = wv >> 5;
  const float* Vb = V32 + (size_t)b * NN * WW;
  const float* Hb = H2 + (size_t)b * NN * 256;
  int m16 = lane & 15, hi = lane >> 4;
  v8f acc = {};
  for (int k0 = 0; k0 < NN; k0 += 4) {
    int kk = k0 + hi * 2;
    v2f a, bb;
    a.x  = Vb[(size_t)kk * WW + it * 16 + m16];
    a.y  = Vb[(size_t)(kk + 1) * WW + it * 16 + m16];
    bb.x = Hb[(size_t)kk * 256 + ct * 16 + m16];
    bb.y = Hb[(size_t)(kk + 1) * 256 + ct * 16 + m16];
    acc = wmma4(a, bb, acc);
  }
  float* Gb = G + ((size_t)b * 32 + it * 16) * 256 + ct * 16;
#pragma unroll
  for (int r = 0; r < 8; ++r) Gb[(size_t)(r + hi * 8) * 256 + m16] = acc[r];
}

// ---------------------------------------------------------------------------
// u[b,k] = sum_c W2psi[k,c]*G[b,i(k),c] + b2psi[k]*sv[b,i(k)],  i(k)=(k>>2)&31
// thread per output: 16384 threads
// ---------------------------------------------------------------------------
__global__ void k_u(const float* __restrict__ W2, const float* __restrict__ b2,
                    const float* __restrict__ G, const float* __restrict__ sv,
                    float* __restrict__ U) {
  int idx = blockIdx.x * blockDim.x + threadIdx.x;
  int k = idx & (NN - 1), b = idx >> 12;
  int i = (k >> 2) & 31;
  const float* g = G + ((size_t)b * 32 + i) * 256;
  const float* w = W2 + (size_t)k * 256;
  float s = 0.f;
  for (int c = 0; c < 256; c += 4)
    s += w[c] * g[c] + w[c + 1] * g[c + 1] + w[c + 2] * g[c + 2] + w[c + 3] * g[c + 3];
  U[idx] = s + b2[k] * sv[b * 32 + i];
}

// ---------------------------------------------------------------------------
// M[b,o,c] = sum_j W2phi[o*128+j, c] * u[b,o*128+j];  c0[b,o] = sum_j b2phi*u
// block per (b,o): grid 128, block 256
// ---------------------------------------------------------------------------
__global__ void k_M(const float* __restrict__ W2, const float* __restrict__ b2,
                    const float* __restrict__ U, float* __restrict__ Mo,
                    float* __restrict__ c0) {
  int b = blockIdx.x >> 5, o = blockIdx.x & 31;
  const float* ub = U + (size_t)b * NN + o * 128;
  __shared__ float us[128];
  __shared__ float cr[128];
  int t = threadIdx.x;
  if (t < 128) { us[t] = ub[t]; cr[t] = us[t] * b2[o * 128 + t]; }
  __syncthreads();
  float s = 0.f;
  for (int j = 0; j < 128; ++j)
    s += us[j] * W2[((size_t)(o * 128 + j)) * 256 + t];
  Mo[((size_t)b * 32 + o) * 256 + t] = s;
  if (t == 0) {
    float c = 0.f;
    for (int j = 0; j < 128; ++j) c += cr[j];
    c0[b * 32 + o] = c;
  }
}

// ---------------------------------------------------------------------------
// x = (h2phi @ M^T + c0)/N + v32 @ wi^T + bi ; also BN partial sums (atomics).
// wave per (b, rt in 0..255, ct in 0..1): 2048 waves; block 256 -> grid 256
// ---------------------------------------------------------------------------
__global__ void k_out1(const float* __restrict__ H2, const float* __restrict__ Mo,
                       const float* __restrict__ c0, const float* __restrict__ V32,
                       const float* __restrict__ Wi, const float* __restrict__ bi,
                       float* __restrict__ X, float* __restrict__ bnsum) {
  int wv   = blockIdx.x * 8 + (threadIdx.x >> 5);
  int lane = threadIdx.x & 31;
  int ct = wv & 1, rt = (wv >> 1) & 255, b = wv >> 9;

  const float* Hr = H2 + ((size_t)b * NN + rt * 16) * 256;
  const float* MT = Mo + (size_t)b * 32 * 256 + (size_t)ct * 16 * 256;
  v8f a1 = {};
  for (int k0 = 0; k0 < 256; k0 += 4)
    a1 = wmma4(frag_a(Hr, 256, k0), frag_bT(MT, 256, k0), a1);

  const float* Vr = V32 + ((size_t)b * NN + rt * 16) * WW;
  v8f a2 = {};
  for (int k0 = 0; k0 < WW; k0 += 4)
    a2 = wmma4(frag_a(Vr, WW, k0), frag_bT(Wi + (size_t)ct * 16 * WW, WW, k0), a2);

  int n = lane & 15, hi = lane >> 4;
  int col = ct * 16 + n;
  const float invN = 1.f / (float)NN;
  float cc = c0[b * 32 + col] * invN;
  float bb = bi[col];
  float* Xr = X + ((size_t)b * NN + rt * 16) * WW + ct * 16;
  float s = 0.f, s2 = 0.f;
#pragma unroll
  for (int r = 0; r < 8; ++r) {
    float x = a1[r] * invN + cc + a2[r] + bb;
    Xr[(size_t)(r + hi * 8) * WW + n] = x;
    s += x; s2 += x * x;
  }
  s  += __shfl_xor(s, 16, 32);
  s2 += __shfl_xor(s2, 16, 32);
  if (lane < 16) {
    atomicAdd(&bnsum[col], s);
    atomicAdd(&bnsum[32 + col], s2);
  }
}

// ---------------------------------------------------------------------------
// BatchNorm (training mode, biased var) + optional ReLU
// ---------------------------------------------------------------------------
__global__ void k_bn(const float* __restrict__ X, const float* __restrict__ bnsum,
                     const float* __restrict__ gamma, const float* __restrict__ beta,
                     float* __restrict__ V32, int doRelu) {
  int idx = blockIdx.x * blockDim.x + threadIdx.x;   // BN*32 elements
  int o = idx & 31;
  const float cnt = (float)BN;
  float mean = bnsum[o] / cnt;
  float var  = bnsum[32 + o] / cnt - mean * mean;
  float y = (X[idx] - mean) * rsqrtf(var + 1e-5f) * gamma[o] + beta[o];
  if (doRelu) y = fmaxf(y, 0.f);
  V32[idx] = y;
}

__global__ void k_zero(float* __restrict__ p) { p[threadIdx.x] = 0.f; }

// ---------------------------------------------------------------------------
// Head: out = relu(v32 @ fc1^T + b1) @ fc2^T + b2.  Wave per row.
// ---------------------------------------------------------------------------
__global__ void k_head(const float* __restrict__ V32,
                       const float* __restrict__ W1, const float* __restrict__ b1,
                       const float* __restrict__ W2, const float* __restrict__ b2,
                       float* __restrict__ out) {
  int row  = blockIdx.x * 8 + (threadIdx.x >> 5);
  int lane = threadIdx.x & 31;
  float vreg = V32[(size_t)row * WW + lane];
  float acc = 0.f;
#pragma unroll
  for (int j = 0; j < 4; ++j) {
    int n = j * 32 + lane;
    float h = b1[n];
    for (int k = 0; k < 32; ++k) h += __shfl(vreg, k, 32) * W1[(size_t)n * WW + k];
    acc += fmaxf(h, 0.f) * W2[n];
  }
  for (int off = 16; off > 0; off >>= 1) acc += __shfl_xor(acc, off, 32);
  if (lane == 0) out[row] = acc + b2[0];
}

// ---------------------------------------------------------------------------
// Workspace layout (floats).  Total ~9.52M floats (~38 MB).
// ---------------------------------------------------------------------------
static constexpr size_t OFF_V32 = 0;
static constexpr size_t OFF_X   = OFF_V32 + (size_t)BN * WW;
static constexpr size_t OFF_H2A = OFF_X + (size_t)BN * WW;          // phi trunk
static constexpr size_t OFF_H2B = OFF_H2A + (size_t)BN * 256;       // psi trunk
static constexpr size_t OFF_G   = OFF_H2B + (size_t)BN * 256;
static constexpr size_t OFF_SV  = OFF_G + (size_t)BB * 32 * 256;
static constexpr size_t OFF_U   = OFF_SV + 128;
static constexpr size_t OFF_M   = OFF_U + (size_t)BB * NN;
static constexpr size_t OFF_C0  = OFF_M + (size_t)BB * 32 * 256;
static constexpr size_t OFF_BN  = OFF_C0 + 128;                     // 64 floats

extern "C" void kernel_launch(void* const* d_in, const int* in_sizes, int n_in,
                              void* d_out, int out_size, void* d_ws, size_t ws_size,
                              hipStream_t stream) {
  (void)in_sizes; (void)n_in; (void)out_size; (void)ws_size;
  auto in = [&](int i) { return (const float*)d_in[i]; };
  float* ws  = (float*)d_ws;
  float* v32 = ws + OFF_V32;
  float* X   = ws + OFF_X;
  float* h2a = ws + OFF_H2A;
  float* h2b = ws + OFF_H2B;
  float* G   = ws + OFF_G;
  float* sv  = ws + OFF_SV;
  float* U   = ws + OFF_U;
  float* Mo  = ws + OFF_M;
  float* c0  = ws + OFF_C0;
  float* bns = ws + OFF_BN;
  const float* v = in(0);
  float* out = (float*)d_out;

  k_fc0<<<256, 256, 0, stream>>>(v, in(1), in(2), v32);

  for (int i = 0; i < 4; ++i) {
    int L = 3 + i * 16;
    k_zero<<<1, 64, 0, stream>>>(bns);
    // phi / psi trunks (56->64->256)
    k_mlp<<<128, 256, 0, stream>>>(v, in(L + 0), in(L + 1), in(L + 2), in(L + 3), h2a);
    k_mlp<<<128, 256, 0, stream>>>(v, in(L + 6), in(L + 7), in(L + 8), in(L + 9), h2b);
    k_sv<<<BB, 1024, 0, stream>>>(v32, sv);
    k_G<<<32, 128, 0, stream>>>(v32, h2b, G);
    k_u<<<64, 256, 0, stream>>>(in(L + 10), in(L + 11), G, sv, U);
    k_M<<<128, 256, 0, stream>>>(in(L + 4), in(L + 5), U, Mo, c0);
    k_out1<<<256, 256, 0, stream>>>(h2a, Mo, c0, v32, in(L + 12), in(L + 13), X, bns);
    k_bn<<<2048, 256, 0, stream>>>(X, bns, in(L + 14), in(L + 15), v32, (i != 3) ? 1 : 0);
  }

  k_head<<<2048, 256, 0, stream>>>(v32, in(67), in(68), in(69), in(70), out);
}